// Generate_dynamic_A_22299470201296
// MI455X (gfx1250) — compile-verified
//
#include <hip/hip_runtime.h>

typedef __attribute__((ext_vector_type(2))) float v2f;
typedef __attribute__((ext_vector_type(8))) float v8f;

#define BATCH 8
#define NN    2048
#define FF    64
#define EPSN  1e-6f

// ---------------------------------------------------------------------------
// Kernel 1: dual layernorm (unbiased std, ddof=1) over last dim F=64.
// One wave32 per row; each lane owns elements {lane, lane+32}.
// ---------------------------------------------------------------------------
__global__ void norm2_kernel(const float* __restrict__ x,
                             const float* __restrict__ a1, const float* __restrict__ b1,
                             const float* __restrict__ a2, const float* __restrict__ b2,
                             float* __restrict__ x1, float* __restrict__ x2) {
    int row  = blockIdx.x * (blockDim.x >> 5) + (threadIdx.x >> 5);
    int lane = threadIdx.x & 31;
    const float* xr = x + (size_t)row * FF;
    float v0 = xr[lane], v1 = xr[lane + 32];
    float s  = v0 + v1;
    float sq = v0 * v0 + v1 * v1;
    #pragma unroll
    for (int off = 16; off > 0; off >>= 1) {
        s  += __shfl_xor(s,  off, 32);
        sq += __shfl_xor(sq, off, 32);
    }
    float mean = s * (1.0f / 64.0f);
    float var  = (sq - 64.0f * mean * mean) * (1.0f / 63.0f);   // unbiased (ddof=1)
    float inv  = 1.0f / (sqrtf(fmaxf(var, 0.0f)) + EPSN);
    float c0 = (v0 - mean) * inv;
    float c1 = (v1 - mean) * inv;
    size_t base = (size_t)row * FF;
    x1[base + lane]      = a1[lane]      * c0 + b1[lane];
    x1[base + lane + 32] = a1[lane + 32] * c1 + b1[lane + 32];
    x2[base + lane]      = a2[lane]      * c0 + b2[lane];
    x2[base + lane + 32] = a2[lane + 32] * c1 + b2[lane + 32];
}

// ---------------------------------------------------------------------------
// Kernel 2: z[b] = conv_w @ x1[b]   ([2048x2048] x [2048x64] -> [2048x64])
// One wave per 16x16 output tile; K=2048 in steps of 4 via wmma_f32_16x16x4_f32.
// fp32 A 16x4 layout: lanes 0-15 hold K={0,1}, lanes 16-31 hold K={2,3}
// -> lane reads contiguous float2 of W at col k + 2*half.
// ---------------------------------------------------------------------------
__global__ void gemm1_kernel(const float* __restrict__ W,
                             const float* __restrict__ x1,
                             float* __restrict__ z) {
    int tile = blockIdx.x;
    int b    = tile / (128 * 4);
    int r    = tile % (128 * 4);
    int tm   = r / 4;            // output row tile (o)
    int tf   = r % 4;            // output col tile (f)
    int lane = threadIdx.x;
    int half = lane >> 4;
    int l15  = lane & 15;

    const float* Wrow = W  + (size_t)(tm * 16 + l15) * NN;
    const float* x1b  = x1 + (size_t)b * NN * FF;
    int col = tf * 16 + l15;     // N index (f)

    v8f acc = {};
    #pragma unroll 4
    for (int k = 0; k < NN; k += 4) {
        int kk = k + 2 * half;
        v2f a = *(const v2f*)(Wrow + kk);                 // W[o][kk], W[o][kk+1]
        v2f bb;
        bb.x = x1b[(size_t)kk * FF + col];                // x1[kk][f]
        bb.y = x1b[(size_t)(kk + 1) * FF + col];          // x1[kk+1][f]
        acc = __builtin_amdgcn_wmma_f32_16x16x4_f32(
            false, a, false, bb, (short)0, acc, false, false);
    }

    float* zb = z + (size_t)b * NN * FF;
    #pragma unroll
    for (int rr = 0; rr < 8; rr++) {
        int o = tm * 16 + rr + 8 * half;                  // C layout: M = rr + 8*half
        zb[(size_t)o * FF + col] = acc[rr];
    }
}

// ---------------------------------------------------------------------------
// Kernel 3: y[b] = relu(z[b] @ x2[b]^T + conv_b)   -> [2048x2048]
// One wave per 32x32 output tile: 4 WMMA accumulators, 2 A + 2 B float2 loads
// feed 4 WMMAs per k-step (halves L2 reads vs 16x16 tiles). B = x2^T so
// B(k,n) = x2[n][k]: fixed column n per lane -> contiguous float2 from x2 row.
// Output stores are non-temporal: y is write-once, keep W/z/x2 resident in L2.
// ---------------------------------------------------------------------------
__global__ void gemm2_kernel(const float* __restrict__ z,
                             const float* __restrict__ x2,
                             const float* __restrict__ bias,
                             float* __restrict__ y) {
    int tile = blockIdx.x;
    int b    = tile / (64 * 64);
    int r    = tile % (64 * 64);
    int to   = r / 64;           // 32-row tile (o)
    int tn   = r % 64;           // 32-col tile (m)
    int lane = threadIdx.x;
    int half = lane >> 4;
    int l15  = lane & 15;

    const float* zb  = z  + (size_t)b * NN * FF;
    const float* x2b = x2 + (size_t)b * NN * FF;
    const float* zrow0 = zb  + (size_t)(to * 32 + l15) * FF;
    const float* zrow1 = zrow0 + 16 * FF;
    const float* xrow0 = x2b + (size_t)(tn * 32 + l15) * FF;
    const float* xrow1 = xrow0 + 16 * FF;

    v8f acc00 = {}, acc01 = {}, acc10 = {}, acc11 = {};
    #pragma unroll
    for (int k = 0; k < FF; k += 4) {
        int kk = k + 2 * half;
        v2f a0 = *(const v2f*)(zrow0 + kk);   // z rows to*32 + [0..15]
        v2f a1 = *(const v2f*)(zrow1 + kk);   // z rows to*32 + [16..31]
        v2f b0 = *(const v2f*)(xrow0 + kk);   // x2 rows tn*32 + [0..15]
        v2f b1 = *(const v2f*)(xrow1 + kk);   // x2 rows tn*32 + [16..31]
        acc00 = __builtin_amdgcn_wmma_f32_16x16x4_f32(false, a0, false, b0, (short)0, acc00, false, false);
        acc01 = __builtin_amdgcn_wmma_f32_16x16x4_f32(false, a0, false, b1, (short)0, acc01, false, false);
        acc10 = __builtin_amdgcn_wmma_f32_16x16x4_f32(false, a1, false, b0, (short)0, acc10, false, false);
        acc11 = __builtin_amdgcn_wmma_f32_16x16x4_f32(false, a1, false, b1, (short)0, acc11, false, false);
    }

    float* yb = y + (size_t)b * NN * NN;
    int m0 = tn * 32 + l15;        // columns for acc*0
    #pragma unroll
    for (int rr = 0; rr < 8; rr++) {
        int o0 = to * 32 + rr + 8 * half;     // rows of acc0*
        int o1 = o0 + 16;                     // rows of acc1*
        float bia0 = bias[o0];
        float bia1 = bias[o1];
        float* p0 = yb + (size_t)o0 * NN + m0;
        float* p1 = yb + (size_t)o1 * NN + m0;
        __builtin_nontemporal_store(fmaxf(acc00[rr] + bia0, 0.0f), p0);
        __builtin_nontemporal_store(fmaxf(acc01[rr] + bia0, 0.0f), p0 + 16);
        __builtin_nontemporal_store(fmaxf(acc10[rr] + bia1, 0.0f), p1);
        __builtin_nontemporal_store(fmaxf(acc11[rr] + bia1, 0.0f), p1 + 16);
    }
}

// ---------------------------------------------------------------------------
extern "C" void kernel_launch(void* const* d_in, const int* in_sizes, int n_in,
                              void* d_out, int out_size, void* d_ws, size_t ws_size,
                              hipStream_t stream) {
    const float* x  = (const float*)d_in[0];   // [8,2048,64]
    const float* a1 = (const float*)d_in[1];   // [64]
    const float* b1 = (const float*)d_in[2];   // [64]
    const float* a2 = (const float*)d_in[3];   // [64]
    const float* b2 = (const float*)d_in[4];   // [64]
    const float* W  = (const float*)d_in[5];   // [2048,2048]
    const float* cb = (const float*)d_in[6];   // [2048]
    float* y = (float*)d_out;                  // [8,2048,2048]

    float* x1 = (float*)d_ws;                          // 8*2048*64 floats (4 MB)
    float* x2 = x1 + (size_t)BATCH * NN * FF;          // 4 MB
    float* z  = x2 + (size_t)BATCH * NN * FF;          // 4 MB

    // 1) dual layernorm: 16384 rows, one wave each, 8 waves per block
    norm2_kernel<<<(BATCH * NN) / 8, 256, 0, stream>>>(x, a1, b1, a2, b2, x1, x2);

    // 2) z = W @ x1 : 8 * 128 * 4 tiles, one wave per tile
    gemm1_kernel<<<BATCH * 128 * 4, 32, 0, stream>>>(W, x1, z);

    // 3) y = relu(z @ x2^T + b) : 8 * 64 * 64 tiles of 32x32, one wave per tile
    gemm2_kernel<<<BATCH * 64 * 64, 32, 0, stream>>>(z, x2, cb, y);
}